// SimpleRNN_85796266705206
// MI455X (gfx1250) — compile-verified
//
#include <hip/hip_runtime.h>
#include <math.h>

// ---------------------------------------------------------------------------
// SimpleRNN forward for MI455X (gfx1250), f32 end-to-end via V_WMMA_F32_16X16X4_F32.
//
// Outputs (concatenated in d_out, all f32):
//   [0] output_tensor (512,64,256)   =  8,388,608
//   [1] input_proj    (64,512,1024)  = 33,554,432
//   [2] tot_rnnhid    (64,512,1024)  = 33,554,432
//   [3] tot_output    (64,512,256)   =  8,388,608
//
// Workspace (d_ws): tanh(h) ping-pong 2*64*1024 f32 (512KB) + 2 u32 barrier.
// ---------------------------------------------------------------------------

typedef float v2f __attribute__((ext_vector_type(2)));
typedef float v8f __attribute__((ext_vector_type(8)));
typedef unsigned v4u __attribute__((ext_vector_type(4)));
typedef int v4i __attribute__((ext_vector_type(4)));
typedef int v8i __attribute__((ext_vector_type(8)));

#define TSQ   512
#define BATCH 64
#define NIN   256
#define HID   1024
#define NOUT  256
#define A_DT  0.05f

#define OFF_IP   ((size_t)TSQ * BATCH * NOUT)                  //  8388608
#define OFF_HID  (OFF_IP + (size_t)BATCH * TSQ * HID)          // 41943040
#define OFF_TOUT (OFF_HID + (size_t)BATCH * TSQ * HID)         // 75497472

#define NBLK_REC 16   // persistent blocks in recurrent kernel (64 cols each)

#if defined(__has_builtin)
#if __has_builtin(__builtin_amdgcn_tanhf)
#define HAVE_HW_TANH 1
#endif
#if __has_builtin(__builtin_amdgcn_tensor_load_to_lds)
#define HAVE_TDM 1
#endif
#endif

__device__ __forceinline__ float fast_tanh(float x) {
#ifdef HAVE_HW_TANH
  return __builtin_amdgcn_tanhf(x);     // v_tanh_f32 (TRANS, co-executes w/ WMMA)
#else
  return tanhf(x);
#endif
}

// ---------------------------------------------------------------------------
// Device-wide sense barrier for the persistent recurrent kernel.
// ---------------------------------------------------------------------------
__device__ __forceinline__ void grid_barrier(volatile unsigned* cnt,
                                             volatile unsigned* gen,
                                             unsigned nblocks) {
  __syncthreads();
  if (threadIdx.x == 0) {
    __threadfence();                       // release prior global writes
    unsigned g = *gen;                     // read generation BEFORE arriving
    unsigned prev = atomicAdd((unsigned*)cnt, 1u);
    if (prev == nblocks - 1u) {
      *cnt = 0u;
      __threadfence();
      atomicAdd((unsigned*)gen, 1u);       // release
    } else {
      while (*gen == g) { __builtin_amdgcn_s_sleep(1); }
    }
    __threadfence();                       // acquire
  }
  __syncthreads();
}

// ---------------------------------------------------------------------------
// Init: zero tanh(h) parity-0 buffer (h0 = 0 -> tanh = 0), reset barrier.
// ---------------------------------------------------------------------------
__global__ void k_init(float* __restrict__ thbuf, unsigned* __restrict__ bar) {
  unsigned i = blockIdx.x * blockDim.x + threadIdx.x;
  if (i < (unsigned)(BATCH * HID)) thbuf[i] = 0.0f;
  if (i == 0u) { bar[0] = 0u; bar[1] = 0u; }
}

// ---------------------------------------------------------------------------
// Kernel 1: input_proj[b,t,h] = sum_i x[t,b,i] * Win[i,h]
// One wave computes a 16x64 C strip (4 accumulators reuse one A fragment).
// Rows r = t*64+b. K = 256.
// ---------------------------------------------------------------------------
__global__ void __launch_bounds__(256)
k_input_proj(const float* __restrict__ x, const float* __restrict__ Win,
             float* __restrict__ ip) {
  const unsigned wave = (blockIdx.x * blockDim.x + threadIdx.x) >> 5;
  const unsigned lane = threadIdx.x & 31u;
  const unsigned tm = wave >> 4;              // 0..2047  (M tile)
  const unsigned c0 = (wave & 15u) * 64u;     // N strip base (0..960)
  const unsigned r0 = tm * 16u;
  const unsigned ln = lane & 15u, hi = lane >> 4, kk = 2u * hi;

  const float* __restrict__ arow = x + (size_t)(r0 + ln) * NIN;
  v8f acc[4] = {};
#pragma unroll 4
  for (int k = 0; k < NIN; k += 4) {
    v2f a = *(const v2f*)(arow + k + kk);
    const float* __restrict__ brow = Win + (size_t)(k + kk) * HID + c0 + ln;
#pragma unroll
    for (int j = 0; j < 4; ++j) {
      v2f b;
      b.x = brow[16 * j];
      b.y = brow[16 * j + HID];
      acc[j] = __builtin_amdgcn_wmma_f32_16x16x4_f32(false, a, false, b,
                                                     (short)0, acc[j],
                                                     false, false);
    }
  }

  // rows r0..r0+15 share t (64 % 16 == 0); scatter into (b,t,h) layout
  const unsigned t  = r0 / BATCH;
  const unsigned b0 = r0 % BATCH;
#pragma unroll
  for (int j = 0; j < 4; ++j) {
#pragma unroll
    for (int i = 0; i < 8; ++i) {
      unsigned bb = b0 + (unsigned)i + 8u * hi;
      ip[(size_t)bb * (TSQ * HID) + (size_t)t * HID + c0 + 16u * j + ln] =
          acc[j][i];
    }
  }
}

// ---------------------------------------------------------------------------
// Kernel 2: persistent recurrent scan.
// 16 blocks x 256 threads. Block g owns Wr[:, g*64 : g*64+64) in LDS (256 KB),
// staged via the Tensor Data Mover when available. Wave w owns a 16x32 h strip
// (2 accumulators, shared A fragment). h lives in registers for all 512 steps;
// tanh(h) ping-pongs through an L2-resident buffer, one grid barrier per step.
// ---------------------------------------------------------------------------
__global__ void __launch_bounds__(256)
k_recurrent(const float* __restrict__ Wr, const float* __restrict__ bias,
            const float* __restrict__ ip, float* __restrict__ hid,
            float* __restrict__ thbuf, unsigned* __restrict__ bar) {
  extern __shared__ float lWr[];                 // [1024][64] = 256 KB
  const unsigned tid = threadIdx.x;
  const unsigned w = tid >> 5, lane = tid & 31u;
  const unsigned c0 = blockIdx.x * 64u;

#ifdef HAVE_TDM
  if (w == 0) {
    // D# descriptor: 2D tile, 4-byte elements, 64 wide x 1024 rows,
    // row stride 1024 elements, dest = LDS offset 0 (dynamic LDS base).
    unsigned long long ga = (unsigned long long)(uintptr_t)(Wr + c0);
    v4u g0;
    g0[0] = 1u;                                   // count=1, user mode
    g0[1] = 0u;                                   // lds_addr (bytes)
    g0[2] = (unsigned)(ga & 0xFFFFFFFFull);       // global_addr[31:0]
    g0[3] = (unsigned)((ga >> 32) & 0x01FFFFFFull) | (2u << 30);  // addr[56:32] | type=2
    v8i g1;
    g1[0] = (int)(2u << 16);                      // data_size=2 (4B), wg_mask=0
    g1[1] = (int)((HID & 0xFFFFu) << 16);         // tensor_dim0[15:0] @ [63:48]
    g1[2] = (int)((HID & 0xFFFFu) << 16);         // tensor_dim1[15:0] @ [95:80]
    g1[3] = (int)(64u << 16);                     // tile_dim0=64 @ [127:112]
    g1[4] = (int)(HID & 0xFFFFu);                 // tile_dim1=1024 @ [143:128]
    g1[5] = (int)HID;                             // tensor_dim0_stride[31:0]
    g1[6] = 0;
    g1[7] = 0;
    v4i gz4 = {};
#if __clang_major__ >= 23
    v8i gz8 = {};
    __builtin_amdgcn_tensor_load_to_lds(g0, g1, gz4, gz4, gz8, 0);
#else
    __builtin_amdgcn_tensor_load_to_lds(g0, g1, gz4, gz4, 0);
#endif
    __builtin_amdgcn_s_wait_tensorcnt(0);
  }
  __syncthreads();
#else
  for (unsigned idx = tid; idx < HID * 64u; idx += 256u) {
    unsigned k = idx >> 6, j = idx & 63u;
    lWr[idx] = Wr[(size_t)k * HID + c0 + j];
  }
  __syncthreads();
#endif

  const unsigned ln = lane & 15u, hi = lane >> 4, kk = 2u * hi;
  const unsigned m0 = (w & 3u) * 16u;            // batch-row tile base
  const unsigned n0 = (w >> 2) * 32u;            // local col strip base (0/32)
  const unsigned gcol = c0 + n0 + ln;            // global hidden column (tile 0)
  const float bval0 = bias[gcol];
  const float bval1 = bias[gcol + 16u];

  float* th0 = thbuf;
  float* th1 = thbuf + BATCH * HID;
  v8f h[2] = {};

  for (int t = 0; t < TSQ; ++t) {
    const float* __restrict__ th = (t & 1) ? th1 : th0;  // tanh(h_t)
    float* __restrict__ thn      = (t & 1) ? th0 : th1;  // tanh(h_{t+1})

    const float* __restrict__ arow = th + (size_t)(m0 + ln) * HID;
    v8f acc[2] = {};
#pragma unroll 8
    for (int k = 0; k < HID; k += 4) {
      v2f a = *(const v2f*)(arow + k + kk);
      const float* __restrict__ brow = lWr + (k + kk) * 64 + n0 + ln;
      v2f b0, b1;
      b0.x = brow[0];   b0.y = brow[64];
      b1.x = brow[16];  b1.y = brow[80];
      acc[0] = __builtin_amdgcn_wmma_f32_16x16x4_f32(false, a, false, b0,
                                                     (short)0, acc[0],
                                                     false, false);
      acc[1] = __builtin_amdgcn_wmma_f32_16x16x4_f32(false, a, false, b1,
                                                     (short)0, acc[1],
                                                     false, false);
    }

#pragma unroll
    for (int j = 0; j < 2; ++j) {
      const float bv = j ? bval1 : bval0;
      const unsigned gc = gcol + 16u * j;
#pragma unroll
      for (int i = 0; i < 8; ++i) {
        unsigned bb = m0 + (unsigned)i + 8u * hi;
        size_t bth = (size_t)bb * (TSQ * HID) + (size_t)t * HID + gc;
        float hn = (1.0f - A_DT) * h[j][i] + A_DT * (acc[j][i] + ip[bth] + bv);
        h[j][i] = hn;
        hid[bth] = hn;                               // tot_rnnhid[b,t,h]
        thn[(size_t)bb * HID + gc] = fast_tanh(hn);  // stage for step t+1
      }
    }
    grid_barrier(bar, bar + 1, NBLK_REC);            // publish tanh(h_{t+1})
  }
}

// ---------------------------------------------------------------------------
// Kernel 3: tot_output = tanh(tot_rnnhid) @ Wout, plus transposed copy.
// One wave computes a 16x64 C strip (4 accumulators share one tanh'd A frag),
// so the 128 MB hid buffer is streamed from HBM only 4x. K = 1024.
// ---------------------------------------------------------------------------
__global__ void __launch_bounds__(256)
k_output(const float* __restrict__ hid, const float* __restrict__ Wout,
         float* __restrict__ toto, float* __restrict__ outT) {
  const unsigned wave = (blockIdx.x * blockDim.x + threadIdx.x) >> 5;
  const unsigned lane = threadIdx.x & 31u;
  const unsigned tm = wave >> 2;              // 0..2047 (M tile)
  const unsigned c0 = (wave & 3u) * 64u;      // N strip base (0..192)
  const unsigned r0 = tm * 16u;
  const unsigned ln = lane & 15u, hi = lane >> 4, kk = 2u * hi;

  const float* __restrict__ arow = hid + (size_t)(r0 + ln) * HID;
  v8f acc[4] = {};
#pragma unroll 4
  for (int k = 0; k < HID; k += 4) {
    if ((k & 63) == 0) __builtin_prefetch(arow + k + 256, 0, 0);
    v2f raw = *(const v2f*)(arow + k + kk);
    v2f a;  a.x = fast_tanh(raw.x);  a.y = fast_tanh(raw.y);
    const float* __restrict__ brow = Wout + (size_t)(k + kk) * NOUT + c0 + ln;
#pragma unroll
    for (int j = 0; j < 4; ++j) {
      v2f b;
      b.x = brow[16 * j];
      b.y = brow[16 * j + NOUT];
      acc[j] = __builtin_amdgcn_wmma_f32_16x16x4_f32(false, a, false, b,
                                                     (short)0, acc[j],
                                                     false, false);
    }
  }

  // rows r0..r0+15 share b (512 % 16 == 0)
  const unsigned bb = r0 / TSQ;
  const unsigned t0 = r0 % TSQ;
#pragma unroll
  for (int j = 0; j < 4; ++j) {
#pragma unroll
    for (int i = 0; i < 8; ++i) {
      unsigned mm = (unsigned)i + 8u * hi;
      unsigned r = r0 + mm, t = t0 + mm;
      float v = acc[j][i];
      toto[(size_t)r * NOUT + c0 + 16u * j + ln] = v;                // (b,t,o)
      outT[((size_t)t * BATCH + bb) * NOUT + c0 + 16u * j + ln] = v; // (t,b,o)
    }
  }
}

// ---------------------------------------------------------------------------
extern "C" void kernel_launch(void* const* d_in, const int* in_sizes, int n_in,
                              void* d_out, int out_size, void* d_ws, size_t ws_size,
                              hipStream_t stream) {
  const float* x    = (const float*)d_in[0];   // (512,64,256)
  const float* Win  = (const float*)d_in[1];   // (256,1024)
  const float* Wr   = (const float*)d_in[2];   // (1024,1024)
  const float* bias = (const float*)d_in[3];   // (1,1024)
  const float* Wout = (const float*)d_in[4];   // (1024,256)

  float* out  = (float*)d_out;
  float* outT = out;              // output_tensor
  float* ip   = out + OFF_IP;     // input_proj
  float* hid  = out + OFF_HID;    // tot_rnnhid
  float* toto = out + OFF_TOUT;   // tot_output

  float* thbuf = (float*)d_ws;                                   // 2*64*1024 f32
  unsigned* bar = (unsigned*)((char*)d_ws +
                              (size_t)2 * BATCH * HID * sizeof(float));

  k_init<<<(BATCH * HID + 255) / 256, 256, 0, stream>>>(thbuf, bar);

  // 2048 M-tiles x 16 N-strips = 32768 waves -> 4096 blocks of 8 waves
  k_input_proj<<<4096, 256, 0, stream>>>(x, Win, ip);

  // persistent: 16 blocks, 256 KB dynamic LDS each (Wr column slice)
  k_recurrent<<<NBLK_REC, 256, HID * 64 * sizeof(float), stream>>>(
      Wr, bias, ip, hid, thbuf, bar);

  // 2048 M-tiles x 4 N-strips = 8192 waves -> 1024 blocks
  k_output<<<1024, 256, 0, stream>>>(hid, Wout, toto, outT);
}